// StSkillHGNN_11055245820282
// MI455X (gfx1250) — compile-verified
//
#include <hip/hip_runtime.h>

typedef __attribute__((ext_vector_type(2))) float v2f;
typedef __attribute__((ext_vector_type(8))) float v8f;

#define D 128
#define NEG_SLOPE 0.2f
#define WT_STRIDE 132   // 128 + 4 pad: conflict-free ds_load_b64 B fragments

// ---------- monotonic float <-> u32 key (for atomic max over floats) ----------
__device__ __forceinline__ unsigned fkey(float f) {
  unsigned b = __float_as_uint(f);
  return (b & 0x80000000u) ? ~b : (b | 0x80000000u);
}
__device__ __forceinline__ float fkey_inv(unsigned k) {
  return __uint_as_float((k & 0x80000000u) ? (k & 0x7fffffffu) : ~k);
}

// ---------- utility: zero a float region ----------
__global__ void k_zero(float* __restrict__ p, long n) {
  long i = (long)blockIdx.x * blockDim.x + threadIdx.x;
  long stride = (long)gridDim.x * blockDim.x;
  for (; i < n; i += stride) p[i] = 0.0f;
}

// ---------- C[M,128] = A[M,128] @ W[128,128], f32 WMMA, M % 16 == 0 ----------
// One wave per 16-row strip: A fragment loaded once per K-chunk, reused across all
// 8 column tiles (A streamed exactly once from HBM). W staged transposed+padded in
// LDS once per block -> inner-loop B fragment is one conflict-free ds_load_b64.
__global__ void k_gemm_wmma(const float* __restrict__ A, const float* __restrict__ W,
                            float* __restrict__ C, int M) {
  extern __shared__ float Wt[];  // [128][WT_STRIDE]: Wt[n*WT_STRIDE + k] = W[k*128 + n]

  // ---- stage W into LDS (transposed). All threads reach the barrier. ----
  {
    const int t = threadIdx.x;  // 0..127: K-row of W
    const float4* Wrow = (const float4*)(W + (long)t * D);
#pragma unroll 8
    for (int n4 = 0; n4 < 32; ++n4) {
      float4 v = Wrow[n4];
      const int n = n4 * 4;
      Wt[(n + 0) * WT_STRIDE + t] = v.x;   // lanes -> consecutive banks: conflict-free
      Wt[(n + 1) * WT_STRIDE + t] = v.y;
      Wt[(n + 2) * WT_STRIDE + t] = v.z;
      Wt[(n + 3) * WT_STRIDE + t] = v.w;
    }
  }
  __syncthreads();

  const int lane = threadIdx.x & 31;
  const int wave = threadIdx.x >> 5;
  const int tm = blockIdx.x * 4 + wave;     // 16-row strip index
  if (tm * 16 >= M) return;                 // wave-uniform exit: EXEC all-ones for WMMA

  const int rm = lane & 15;
  const int kh = (lane >> 4) << 1;          // 0 or 2: K-pair of this half-wave
  const float* Arow = A + (long)(tm * 16 + rm) * D;

  v8f acc[8] = {};
#pragma unroll 2
  for (int kc = 0; kc < D; kc += 4) {
    const int kk = kc + kh;
    // A 16x4 fragment: lane = row (lane&15), K = kk,kk+1  (8B-aligned b64 load)
    v2f a = *(const v2f*)(Arow + kk);
#pragma unroll
    for (int tn = 0; tn < 8; ++tn) {
      // B 4x16 fragment: lane = col (lane&15), K = kk,kk+1 from transposed LDS
      v2f b = *(const v2f*)(Wt + (long)(tn * 16 + rm) * WT_STRIDE + kk);
      acc[tn] = __builtin_amdgcn_wmma_f32_16x16x4_f32(false, a, false, b, (short)0,
                                                      acc[tn], false, false);
    }
  }

  // C/D layout: VGPR i -> row i (lanes 0-15) / row 8+i (lanes 16-31), col = lane&15
  const int rowbase = tm * 16 + ((lane >> 4) << 3);
#pragma unroll
  for (int tn = 0; tn < 8; ++tn) {
    const int cn = tn * 16 + rm;
#pragma unroll
    for (int i = 0; i < 8; ++i)
      C[(long)(rowbase + i) * D + cn] = acc[tn][i];
  }
}

// ---------- per-row dual dot products: a_s[i] = h[i]·att_s, a_d[i] = h[i]·att_d ----------
__global__ void k_dots(const float* __restrict__ h, const float* __restrict__ att_s,
                       const float* __restrict__ att_d, float* __restrict__ a_s,
                       float* __restrict__ a_d, int Nn) {
  int row = (int)(((long)blockIdx.x * blockDim.x + threadIdx.x) >> 5);
  int lane = threadIdx.x & 31;
  if (row >= Nn) return;
  float4 hv = ((const float4*)(h + (long)row * D))[lane];
  float4 sv = ((const float4*)att_s)[lane];
  float4 dv = ((const float4*)att_d)[lane];
  float s = hv.x * sv.x + hv.y * sv.y + hv.z * sv.z + hv.w * sv.w;
  float d = hv.x * dv.x + hv.y * dv.y + hv.z * dv.z + hv.w * dv.w;
#pragma unroll
  for (int off = 16; off; off >>= 1) {
    s += __shfl_xor(s, off, 32);
    d += __shfl_xor(d, off, 32);
  }
  if (lane == 0) { a_s[row] = s; a_d[row] = d; }
}

// ---------- GAT pass 1: per-dst max of leaky-relu logits (self loops appended) ----------
__global__ void k_gat_max(const int* __restrict__ src, const int* __restrict__ dst,
                          int nE, int Nn, const float* __restrict__ a_s,
                          const float* __restrict__ a_d, unsigned* __restrict__ emax) {
  int e = (int)((long)blockIdx.x * blockDim.x + threadIdx.x);
  if (e >= nE + Nn) return;
  int s = (e < nE) ? src[e] : (e - nE);
  int t = (e < nE) ? dst[e] : (e - nE);
  float x = a_s[s] + a_d[t];
  float el = (x > 0.0f) ? x : NEG_SLOPE * x;
  atomicMax(emax + t, fkey(el));
}

// ---------- GAT pass 2: ex = exp(e - max[dst]); denom[dst] += ex ----------
__global__ void k_gat_exp(const int* __restrict__ src, const int* __restrict__ dst,
                          int nE, int Nn, const float* __restrict__ a_s,
                          const float* __restrict__ a_d, const unsigned* __restrict__ emax,
                          float* __restrict__ ex, float* __restrict__ denom) {
  int e = (int)((long)blockIdx.x * blockDim.x + threadIdx.x);
  if (e >= nE + Nn) return;
  int s = (e < nE) ? src[e] : (e - nE);
  int t = (e < nE) ? dst[e] : (e - nE);
  float x = a_s[s] + a_d[t];
  float el = (x > 0.0f) ? x : NEG_SLOPE * x;
  float v = __expf(el - fkey_inv(emax[t]));
  ex[e] = v;
  atomicAdd(denom + t, v);
}

// ---------- weighted row scatter: out[dst] += w * x[src]   (wave per edge) ----------
__global__ void k_scatter(const int* __restrict__ src, const int* __restrict__ dst,
                          int nE, int nLoop, const float* __restrict__ wgt,
                          const float* __restrict__ x, float* __restrict__ out) {
  int e = (int)(((long)blockIdx.x * blockDim.x + threadIdx.x) >> 5);
  int lane = threadIdx.x & 31;
  if (e >= nE + nLoop) return;
  int s = (e < nE) ? src[e] : (e - nE);
  int t = (e < nE) ? dst[e] : (e - nE);
  float w = wgt ? wgt[e] : 1.0f;
  float4 hv = ((const float4*)(x + (long)s * D))[lane];
  float* op = out + (long)t * D + lane * 4;
  atomicAdd(op + 0, w * hv.x);
  atomicAdd(op + 1, w * hv.y);
  atomicAdd(op + 2, w * hv.z);
  atomicAdd(op + 3, w * hv.w);
}

// ---------- SAGE degree count ----------
__global__ void k_deg(const int* __restrict__ dst, int nE, float* __restrict__ degv) {
  int e = (int)((long)blockIdx.x * blockDim.x + threadIdx.x);
  if (e < nE) atomicAdd(degv + dst[e], 1.0f);
}

// ---------- gather queried rows: g1 = agg[s]/max(deg,1), g2 = emb[s] ----------
__global__ void k_gather(const int* __restrict__ sIdx, int S_, const float* __restrict__ agg,
                         const float* __restrict__ degv, const float* __restrict__ emb,
                         float* __restrict__ g1, float* __restrict__ g2) {
  int j = (int)(((long)blockIdx.x * blockDim.x + threadIdx.x) >> 5);
  int lane = threadIdx.x & 31;
  if (j >= S_) return;
  int i = sIdx[j];
  float inv = 1.0f / fmaxf(degv[i], 1.0f);
  float4 av = ((const float4*)(agg + (long)i * D))[lane];
  float4 ev = ((const float4*)(emb + (long)i * D))[lane];
  av.x *= inv; av.y *= inv; av.z *= inv; av.w *= inv;
  ((float4*)(g1 + (long)j * D))[lane] = av;
  ((float4*)(g2 + (long)j * D))[lane] = ev;
}

// ---------- final combine: normalize GAT sums, add biases, mean over 3 relations ----------
__global__ void k_combine(const int* __restrict__ sIdx, int S_,
                          const float* __restrict__ out_p, const float* __restrict__ denom_p,
                          const float* __restrict__ bp,
                          const float* __restrict__ out_c, const float* __restrict__ denom_c,
                          const float* __restrict__ bc,
                          const float* __restrict__ tA, const float* __restrict__ bl,
                          const float* __restrict__ tB, float* __restrict__ out) {
  int j = (int)(((long)blockIdx.x * blockDim.x + threadIdx.x) >> 5);
  int lane = threadIdx.x & 31;
  if (j >= S_) return;
  int i = sIdx[j];
  float ip = 1.0f / (denom_p[i] + 1e-16f);
  float ic = 1.0f / (denom_c[i] + 1e-16f);
  float4 op = ((const float4*)(out_p + (long)i * D))[lane];
  float4 oc = ((const float4*)(out_c + (long)i * D))[lane];
  float4 ta = ((const float4*)(tA + (long)j * D))[lane];
  float4 tb = ((const float4*)(tB + (long)j * D))[lane];
  float4 vbp = ((const float4*)bp)[lane];
  float4 vbc = ((const float4*)bc)[lane];
  float4 vbl = ((const float4*)bl)[lane];
  const float third = 1.0f / 3.0f;
  float4 r;
  r.x = (op.x * ip + vbp.x + oc.x * ic + vbc.x + ta.x + vbl.x + tb.x) * third;
  r.y = (op.y * ip + vbp.y + oc.y * ic + vbc.y + ta.y + vbl.y + tb.y) * third;
  r.z = (op.z * ip + vbp.z + oc.z * ic + vbc.z + ta.z + vbl.z + tb.z) * third;
  r.w = (op.w * ip + vbp.w + oc.w * ic + vbc.w + ta.w + vbl.w + tb.w) * third;
  ((float4*)(out + (long)j * D))[lane] = r;
}

extern "C" void kernel_launch(void* const* d_in, const int* in_sizes, int n_in,
                              void* d_out, int out_size, void* d_ws, size_t ws_size,
                              hipStream_t stream) {
  const int* sIdx = (const int*)d_in[0];
  // d_in[1], d_in[2] = t_s, t_e (unused by the math)
  const int* ei_p = (const int*)d_in[3];
  const int* ei_c = (const int*)d_in[4];
  const int* ei_r = (const int*)d_in[5];
  const float* emb = (const float*)d_in[6];
  const float* Wp = (const float*)d_in[7];
  const float* asp = (const float*)d_in[8];
  const float* adp = (const float*)d_in[9];
  const float* bp  = (const float*)d_in[10];
  const float* Wc = (const float*)d_in[11];
  const float* asc = (const float*)d_in[12];
  const float* adc = (const float*)d_in[13];
  const float* bc  = (const float*)d_in[14];
  const float* Wl = (const float*)d_in[15];
  const float* bl  = (const float*)d_in[16];
  const float* Wr = (const float*)d_in[17];

  const int S_ = in_sizes[0];
  const int E_ = in_sizes[3] / 2;
  const int Nn = in_sizes[6] / D;

  // ----- workspace layout (floats) -----
  float* ws = (float*)d_ws;
  long off = 0;
  float* h_p = ws + off;            off += (long)Nn * D;
  float* h_c = ws + off;            off += (long)Nn * D;
  long zero_begin = off;
  float* out_p = ws + off;          off += (long)Nn * D;
  float* out_c = ws + off;          off += (long)Nn * D;
  float* agg   = ws + off;          off += (long)Nn * D;
  float* denom_p = ws + off;        off += Nn;
  float* denom_c = ws + off;        off += Nn;
  float* degv    = ws + off;        off += Nn;
  unsigned* emax_p = (unsigned*)(ws + off); off += Nn;
  unsigned* emax_c = (unsigned*)(ws + off); off += Nn;
  long zero_count = off - zero_begin;
  float* as_p = ws + off;           off += Nn;
  float* ad_p = ws + off;           off += Nn;
  float* as_c = ws + off;           off += Nn;
  float* ad_c = ws + off;           off += Nn;
  float* ex_p = ws + off;           off += (long)(E_ + Nn);
  float* ex_c = ws + off;           off += (long)(E_ + Nn);
  float* g1 = ws + off;             off += (long)S_ * D;
  float* g2 = ws + off;             off += (long)S_ * D;
  float* tA = ws + off;             off += (long)S_ * D;
  float* tB = ws + off;             off += (long)S_ * D;

  // 1) zero accumulators (float 0.0 == u32 key 0 == identity for our atomic max)
  k_zero<<<4096, 256, 0, stream>>>(ws + zero_begin, zero_count);

  // 2) h = emb @ W for both GAT relations (f32 WMMA, LDS-staged W, A streamed once)
  const size_t lds_bytes = (size_t)D * WT_STRIDE * sizeof(float);  // 67.6 KB
  int gemm_blocks_n = (Nn / 16 + 3) / 4;
  k_gemm_wmma<<<gemm_blocks_n, 128, lds_bytes, stream>>>(emb, Wp, h_p, Nn);
  k_gemm_wmma<<<gemm_blocks_n, 128, lds_bytes, stream>>>(emb, Wc, h_c, Nn);

  // 3) attention logit dot products
  int dot_blocks = (int)(((long)Nn * 32 + 255) / 256);
  k_dots<<<dot_blocks, 256, 0, stream>>>(h_p, asp, adp, as_p, ad_p, Nn);
  k_dots<<<dot_blocks, 256, 0, stream>>>(h_c, asc, adc, as_c, ad_c, Nn);

  // 4) GAT softmax passes (edges + N self-loops)
  int edge_blocks = (E_ + Nn + 255) / 256;
  k_gat_max<<<edge_blocks, 256, 0, stream>>>(ei_p, ei_p + E_, E_, Nn, as_p, ad_p, emax_p);
  k_gat_max<<<edge_blocks, 256, 0, stream>>>(ei_c, ei_c + E_, E_, Nn, as_c, ad_c, emax_c);
  k_gat_exp<<<edge_blocks, 256, 0, stream>>>(ei_p, ei_p + E_, E_, Nn, as_p, ad_p, emax_p, ex_p, denom_p);
  k_gat_exp<<<edge_blocks, 256, 0, stream>>>(ei_c, ei_c + E_, E_, Nn, as_c, ad_c, emax_c, ex_c, denom_c);

  // 5) weighted scatters (wave per edge, float4 atomics)
  int scat_blocks = (int)((((long)(E_ + Nn)) * 32 + 255) / 256);
  k_scatter<<<scat_blocks, 256, 0, stream>>>(ei_p, ei_p + E_, E_, Nn, ex_p, h_p, out_p);
  k_scatter<<<scat_blocks, 256, 0, stream>>>(ei_c, ei_c + E_, E_, Nn, ex_c, h_c, out_c);

  // 6) SAGE mean aggregation (no self loops)
  int deg_blocks = (E_ + 255) / 256;
  k_deg<<<deg_blocks, 256, 0, stream>>>(ei_r + E_, E_, degv);
  int scat_blocks_r = (int)((((long)E_) * 32 + 255) / 256);
  k_scatter<<<scat_blocks_r, 256, 0, stream>>>(ei_r, ei_r + E_, E_, 0, nullptr, emb, agg);

  // 7) gather queried rows, small WMMA GEMMs for SAGE linear terms
  int q_blocks = (int)(((long)S_ * 32 + 255) / 256);
  k_gather<<<q_blocks, 256, 0, stream>>>(sIdx, S_, agg, degv, emb, g1, g2);
  int gemm_blocks_s = (S_ / 16 + 3) / 4;
  k_gemm_wmma<<<gemm_blocks_s, 128, lds_bytes, stream>>>(g1, Wl, tA, S_);
  k_gemm_wmma<<<gemm_blocks_s, 128, lds_bytes, stream>>>(g2, Wr, tB, S_);

  // 8) combine: (GAT_p + GAT_c + SAGE)/3 at queried rows
  k_combine<<<q_blocks, 256, 0, stream>>>(sIdx, S_, out_p, denom_p, bp, out_c, denom_c, bc,
                                          tA, bl, tB, (float*)d_out);
}